// Group_iSogCLR_Loss_22643067584623
// MI455X (gfx1250) — compile-verified
//
#include <hip/hip_runtime.h>
#include <math.h>

typedef __attribute__((ext_vector_type(2))) float v2f;
typedef __attribute__((ext_vector_type(8))) float v8f;

constexpr int   kB = 4096;
constexpr int   kD = 256;
constexpr int   kG = 100;
constexpr float kAlpha = 0.5f;
constexpr float kGamma = 0.8f;
constexpr float kRho   = 8.0f;
constexpr float kLam   = 1.0f;
constexpr float kEtaP  = 0.01f;
constexpr float kEps   = 1e-14f;
constexpr float kClip  = 5.0f;

// ---------------------------------------------------------------------------
// K1: L2-normalize both embedding matrices. One wave32 per row.
// ---------------------------------------------------------------------------
__global__ void l2norm_kernel(const float* __restrict__ zis,
                              const float* __restrict__ zjs,
                              float* __restrict__ zin,
                              float* __restrict__ zjn) {
  int wave = (blockIdx.x * blockDim.x + threadIdx.x) >> 5;
  int lane = threadIdx.x & 31;
  const float* src;
  float* dst;
  if (wave < kB) {
    src = zis + (size_t)wave * kD;
    dst = zin + (size_t)wave * kD;
  } else {
    int r = wave - kB;
    src = zjs + (size_t)r * kD;
    dst = zjn + (size_t)r * kD;
  }
  float ss = 0.f;
  for (int k = lane; k < kD; k += 32) {
    float x = src[k];
    ss += x * x;
  }
#pragma unroll
  for (int m = 16; m >= 1; m >>= 1) ss += __shfl_xor(ss, m, 32);
  float inv = 1.0f / fmaxf(sqrtf(ss), 1e-12f);
  for (int k = lane; k < kD; k += 32) dst[k] = src[k] * inv;
}

// ---------------------------------------------------------------------------
// K2: diag[i] = dot(zin[i], zjn[i]).  One wave32 per row.
// ---------------------------------------------------------------------------
__global__ void diag_kernel(const float* __restrict__ zin,
                            const float* __restrict__ zjn,
                            float* __restrict__ diag) {
  int row  = (blockIdx.x * blockDim.x + threadIdx.x) >> 5;
  int lane = threadIdx.x & 31;
  float s = 0.f;
  const float* a = zin + (size_t)row * kD;
  const float* b = zjn + (size_t)row * kD;
  for (int k = lane; k < kD; k += 32) s += a[k] * b[k];
#pragma unroll
  for (int m = 16; m >= 1; m >>= 1) s += __shfl_xor(s, m, 32);
  if (lane == 0) diag[row] = s;
}

// ---------------------------------------------------------------------------
// 16x16 sim tile via V_WMMA_F32_16X16X4_F32, K-loop over D=256.
// A(16x4 f32): lane = m + 16*(k>>1), vgpr = k&1  -> per-lane float2 load.
// B(4x16 f32, B[k,n]=Brow[n*D+k]): mirrored layout -> same float2 pattern.
// ---------------------------------------------------------------------------
__device__ __forceinline__ v8f sim_tile(const float* __restrict__ aBase,
                                        const float* __restrict__ bBase) {
  v8f acc = {};
#pragma unroll 8
  for (int k0 = 0; k0 < kD; k0 += 4) {
    v2f a = *reinterpret_cast<const v2f*>(aBase + k0);
    v2f b = *reinterpret_cast<const v2f*>(bBase + k0);
    acc = __builtin_amdgcn_wmma_f32_16x16x4_f32(false, a, false, b,
                                                (short)0, acc, false, false);
  }
  return acc;
}

// ---------------------------------------------------------------------------
// K3/K4: fused pass. For a 16-row block of (Arow @ Brow^T):
//   sweep 1: b_new[row] = max(max_col (sim - diag_row)/tau_row, b_old)
//   sweep 2: g[row] = sum exp(idt - b_new) (mask diag),
//            t[row] = sum exp(idt - b_new) * (sim - diag_row)
// Image pass: Arow=zin, Brow=zjn.  Text pass: Arow=zjn, Brow=zin (transposed).
// ---------------------------------------------------------------------------
__global__ __launch_bounds__(32) void pass_kernel(
    const float* __restrict__ Arow, const float* __restrict__ Brow,
    const float* __restrict__ diag, const float* __restrict__ taus,
    const float* __restrict__ b_state, const int* __restrict__ ids,
    float* __restrict__ out_bnew, float* __restrict__ out_g,
    float* __restrict__ out_t) {
  const int lane = threadIdx.x & 31;
  const int h    = lane >> 4;   // lane half: rows v+8h live here
  const int ln   = lane & 15;   // column within tile / row for A loads
  const int r0   = blockIdx.x * 16;

  float diag_m[8], invtau[8], bold[8];
#pragma unroll
  for (int v = 0; v < 8; ++v) {
    int row   = r0 + v + 8 * h;
    diag_m[v] = diag[row];
    int id    = ids[row];
    invtau[v] = 1.0f / taus[id];
    bold[v]   = b_state[id];
  }

  const float* aBase = Arow + (size_t)(r0 + ln) * kD + 2 * h;

  // ---- sweep 1: row maxima of idt ----
  float pm[8];
#pragma unroll
  for (int v = 0; v < 8; ++v) pm[v] = -3.4e38f;

  for (int c0 = 0; c0 < kB; c0 += 16) {
    const float* bBase = Brow + (size_t)(c0 + ln) * kD + 2 * h;
    v8f acc = sim_tile(aBase, bBase);
#pragma unroll
    for (int v = 0; v < 8; ++v) {
      float idt = (acc[v] - diag_m[v]) * invtau[v];
      pm[v] = fmaxf(pm[v], idt);
    }
  }
#pragma unroll
  for (int v = 0; v < 8; ++v) {
#pragma unroll
    for (int m = 1; m < 16; m <<= 1)
      pm[v] = fmaxf(pm[v], __shfl_xor(pm[v], m, 32));
    pm[v] = fmaxf(pm[v], bold[v]);  // b_new (includes old running max)
  }

  // ---- sweep 2: masked exp sums ----
  float gs[8], ts[8];
#pragma unroll
  for (int v = 0; v < 8; ++v) { gs[v] = 0.f; ts[v] = 0.f; }

  for (int c0 = 0; c0 < kB; c0 += 16) {
    const float* bBase = Brow + (size_t)(c0 + ln) * kD + 2 * h;
    v8f acc = sim_tile(aBase, bBase);
#pragma unroll
    for (int v = 0; v < 8; ++v) {
      int i = r0 + v + 8 * h;
      int j = c0 + ln;
      float dv = acc[v] - diag_m[v];
      float e  = (i != j) ? __expf(dv * invtau[v] - pm[v]) : 0.0f;
      gs[v] += e;
      ts[v] += e * dv;
    }
  }
#pragma unroll
  for (int v = 0; v < 8; ++v) {
#pragma unroll
    for (int m = 1; m < 16; m <<= 1) {
      gs[v] += __shfl_xor(gs[v], m, 32);
      ts[v] += __shfl_xor(ts[v], m, 32);
    }
  }
  if (ln == 0) {
#pragma unroll
    for (int v = 0; v < 8; ++v) {
      int row = r0 + v + 8 * h;
      out_bnew[row] = pm[v];
      out_g[row]    = gs[v];
      out_t[row]    = ts[v];
    }
  }
}

// ---------------------------------------------------------------------------
// K5/K6: per-row finalize: s_new, F, per-row loss.
// ---------------------------------------------------------------------------
__global__ void perrow_kernel(const float* __restrict__ bnew,
                              const float* __restrict__ g,
                              const float* __restrict__ t,
                              const float* __restrict__ s_state,
                              const float* __restrict__ b_state,
                              const float* __restrict__ taus,
                              const int* __restrict__ ids,
                              const int* __restrict__ ginfo,
                              const float* __restrict__ p,
                              float* __restrict__ F_out,
                              float* __restrict__ loss_out) {
  int i = blockIdx.x * blockDim.x + threadIdx.x;
  if (i >= kB) return;
  int id      = ids[i];
  float bn    = bnew[i];
  float s_new = (1.0f - kGamma) * s_state[id] * __expf(b_state[id] - bn) +
                kGamma * g[i];
  float s_c = fmaxf(s_new, kEps);
  float tau = taus[id];
  F_out[i]  = tau * (logf(s_c) + bn + kRho);
  float gw  = (float)kG * p[ginfo[id]];
  loss_out[i] = gw * t[i] / s_c;
}

// ---------------------------------------------------------------------------
// K7: single-block deterministic finalize: segment means, z/p updates, loss.
// ---------------------------------------------------------------------------
__global__ void finalize_kernel(const float* __restrict__ FI,
                                const float* __restrict__ FT,
                                const float* __restrict__ lossI,
                                const float* __restrict__ lossT,
                                const int* __restrict__ ids,
                                const int* __restrict__ gI,
                                const int* __restrict__ gT,
                                const float* __restrict__ zI,
                                const float* __restrict__ zT,
                                const float* __restrict__ pI,
                                const float* __restrict__ pT,
                                float* __restrict__ out) {
  __shared__ float red[256];
  __shared__ float npI[kG], npT[kG], nsum[2];
  int t = threadIdx.x;

  // deterministic tree-reduce of the two loss means
  float lI = 0.f, lT = 0.f;
  for (int i = t; i < kB; i += 256) { lI += lossI[i]; lT += lossT[i]; }
  red[t] = lI;
  __syncthreads();
  for (int s = 128; s > 0; s >>= 1) {
    if (t < s) red[t] += red[t + s];
    __syncthreads();
  }
  float lossSumI = red[0];
  __syncthreads();
  red[t] = lT;
  __syncthreads();
  for (int s = 128; s > 0; s >>= 1) {
    if (t < s) red[t] += red[t + s];
    __syncthreads();
  }
  float lossSumT = red[0];
  __syncthreads();

  // per-group sums (thread t owns group t) -> z, p updates
  if (t < kG) {
    float c = 0.f, f = 0.f;
    for (int i = 0; i < kB; ++i) {
      int id = ids[i];
      if (gI[id] == t) { c += 1.0f; f += FI[i]; }
    }
    float gp = f / fmaxf(c, 1.0f);
    float zn = (1.0f - kGamma) * zI[t] + kGamma * gp;
    float hp = -kLam * logf(pI[t] + kEps) - kLam;
    float w  = fminf(fmaxf(zn + hp, -kClip), kClip);
    npI[t]   = pI[t] * __expf(2.0f * kEtaP * w);

    c = 0.f; f = 0.f;
    for (int i = 0; i < kB; ++i) {
      int id = ids[i];
      if (gT[id] == t) { c += 1.0f; f += FT[i]; }
    }
    gp = f / fmaxf(c, 1.0f);
    zn = (1.0f - kGamma) * zT[t] + kGamma * gp;
    hp = -kLam * logf(pT[t] + kEps) - kLam;
    w  = fminf(fmaxf(zn + hp, -kClip), kClip);
    npT[t] = pT[t] * __expf(2.0f * kEtaP * w);
  }
  __syncthreads();
  if (t == 0) {
    float sI = 0.f, sT = 0.f;
    for (int g = 0; g < kG; ++g) { sI += npI[g]; sT += npT[g]; }
    nsum[0] = sI;
    nsum[1] = sT;
    out[0]  = kAlpha * (lossSumI / (float)kB) +
              (1.0f - kAlpha) * (lossSumT / (float)kB);
  }
  __syncthreads();
  if (t < kG) {
    out[1 + t]      = npI[t] / nsum[0];
    out[1 + kG + t] = npT[t] / nsum[1];
  }
}

// ---------------------------------------------------------------------------
extern "C" void kernel_launch(void* const* d_in, const int* in_sizes, int n_in,
                              void* d_out, int out_size, void* d_ws,
                              size_t ws_size, hipStream_t stream) {
  const float* zis    = (const float*)d_in[0];
  const float* zjs    = (const float*)d_in[1];
  const float* s_I    = (const float*)d_in[2];
  const float* s_T    = (const float*)d_in[3];
  const float* b_I    = (const float*)d_in[4];
  const float* b_T    = (const float*)d_in[5];
  const float* z_I    = (const float*)d_in[6];
  const float* z_T    = (const float*)d_in[7];
  const float* p_I    = (const float*)d_in[8];
  const float* p_T    = (const float*)d_in[9];
  const float* taus_I = (const float*)d_in[10];
  const float* taus_T = (const float*)d_in[11];
  const int*   ids    = (const int*)d_in[12];
  const int*   gI     = (const int*)d_in[13];
  const int*   gT     = (const int*)d_in[14];
  float* out = (float*)d_out;

  float* ws  = (float*)d_ws;
  float* zin  = ws; ws += (size_t)kB * kD;
  float* zjn  = ws; ws += (size_t)kB * kD;
  float* diag = ws; ws += kB;
  float* bnI  = ws; ws += kB;
  float* gIr  = ws; ws += kB;
  float* tIr  = ws; ws += kB;
  float* bnT  = ws; ws += kB;
  float* gTr  = ws; ws += kB;
  float* tTr  = ws; ws += kB;
  float* FI   = ws; ws += kB;
  float* FT   = ws; ws += kB;
  float* lsI  = ws; ws += kB;
  float* lsT  = ws; ws += kB;

  l2norm_kernel<<<(2 * kB) / 8, 256, 0, stream>>>(zis, zjs, zin, zjn);
  diag_kernel<<<kB / 8, 256, 0, stream>>>(zin, zjn, diag);

  // image pass: rows of sim;  text pass: rows of sim^T (swap operands)
  pass_kernel<<<kB / 16, 32, 0, stream>>>(zin, zjn, diag, taus_I, b_I, ids,
                                          bnI, gIr, tIr);
  pass_kernel<<<kB / 16, 32, 0, stream>>>(zjn, zin, diag, taus_T, b_T, ids,
                                          bnT, gTr, tTr);

  perrow_kernel<<<kB / 256, 256, 0, stream>>>(bnI, gIr, tIr, s_I, b_I, taus_I,
                                              ids, gI, p_I, FI, lsI);
  perrow_kernel<<<kB / 256, 256, 0, stream>>>(bnT, gTr, tTr, s_T, b_T, taus_T,
                                              ids, gT, p_T, FT, lsT);

  finalize_kernel<<<1, 256, 0, stream>>>(FI, FT, lsI, lsT, ids, gI, gT, z_I,
                                         z_T, p_I, p_T, out);
}